// WsiNet_85392539779766
// MI455X (gfx1250) — compile-verified
//
#include <hip/hip_runtime.h>
#include <math.h>

// ---------------------------------------------------------------------------
// WsiNet (MIL attention pooling + 2-class head) for MI455X / gfx1250, wave32.
//
//   B=8, N=4096, D=2048, DL=64, NCLS=2
//   out = [cls_probs (B*2) | soft_assign (B*N)]   (float32)
//
// Schedule: single pass over x (256MB, HBM-bound floor ~11us @ 23.3TB/s):
//   K1: per 16-row tile, WMMA f32 16x16x4 accumulates x@V (16x64),
//       fuses tanh+(.@W)/8 -> dis, and fuses per-patch x.fc_w dots -> dotc.
//       V/fc_w chunks double-buffered into LDS via async global->LDS DMA
//       (GLOBAL_LOAD_ASYNC_TO_LDS_B32, ASYNCcnt) overlapped with compute.
//   K2: masked softmax over N per bag -> soft_assign (d_out+16).
//   K3: logits[b,c] = sum_n s[b,n]*dotc[b,n,c] + fc_b[c]; 2-class softmax.
// ---------------------------------------------------------------------------

typedef __attribute__((ext_vector_type(2))) float v2f;
typedef __attribute__((ext_vector_type(8))) float v8f;

#define B_    8
#define N_    4096
#define D_    2048
#define DL_   64
#define NCLS_ 2
#define CK    64          // K-chunk staged in LDS
#define CKP   (CK + 2)    // padded stride (even -> b64-aligned, conflict-free)
#define NCHUNK (D_ / CK)  // 32

// Async global->LDS copy of one V chunk (K-transposed) + one fc_w chunk.
// Per-lane LDS destination addresses let us do the transpose in the DMA.
__device__ __forceinline__ void stage_chunk_async(
    const float* __restrict__ V, const float* __restrict__ fcw,
    float* ldsVbuf, float* ldsFbuf, int kc, int tid)
{
    #pragma unroll
    for (int j = 0; j < (DL_ * CK) / 256; ++j) {
        int idx = j * 256 + tid;
        int kk  = idx >> 6;     // 0..63
        int col = idx & 63;
        const float* g = V + (kc + kk) * DL_ + col;          // coalesced reads
        unsigned l = (unsigned)(uintptr_t)(ldsVbuf + col * CKP + kk);
        asm volatile("global_load_async_to_lds_b32 %0, %1, off"
                     :: "v"(l), "v"(g) : "memory");
    }
    if (tid < NCLS_ * CK) {
        int c = tid >> 6, kk = tid & 63;
        const float* g = fcw + c * D_ + kc + kk;
        unsigned l = (unsigned)(uintptr_t)(ldsFbuf + c * CK + kk);
        asm volatile("global_load_async_to_lds_b32 %0, %1, off"
                     :: "v"(l), "v"(g) : "memory");
    }
}

__device__ __forceinline__ void wait_async_all()
{
    asm volatile("s_wait_asynccnt 0x0" ::: "memory");
}

// ---------------------------------------------------------------------------
// Kernel 1: scores + fused per-patch class dots.
// Block = 256 threads = 8 waves; wave owns 16 rows; block owns 128 rows.
// Grid = (B*N)/128 = 256 blocks. 4096 % 128 == 0 so a block never crosses bags.
// ---------------------------------------------------------------------------
__global__ __launch_bounds__(256) void wsinet_scores(
    const float* __restrict__ x,    // [B*N, D]
    const float* __restrict__ V,    // [D, DL]
    const float* __restrict__ W,    // [DL]
    const float* __restrict__ fcw,  // [NCLS, D]
    float* __restrict__ dis,        // [B*N]
    float* __restrict__ dotc)       // [B*N, NCLS]
{
    __shared__ float ldsV[2][DL_ * CKP];   // V chunks, K-transposed: [col][k]
    __shared__ float ldsF[2][NCLS_ * CK];  // fc_w chunks: [c][k]

    const int tid  = threadIdx.x;
    const int lane = tid & 31;
    const int wave = tid >> 5;
    const int h    = lane >> 4;    // K-half within WMMA fragment (0/1)
    const int m    = lane & 15;    // row-in-tile (A) / col-in-tile (B)
    const long rowbase = (long)blockIdx.x * 128 + (long)wave * 16;

    v8f acc0 = {}, acc1 = {}, acc2 = {}, acc3 = {};   // 16x64 C tile
    float dp0 = 0.0f, dp1 = 0.0f;                     // fused fc_w dots

    const float* xrow = x + (rowbase + m) * (long)D_;

    // prologue: stage chunk 0, wait, barrier
    stage_chunk_async(V, fcw, ldsV[0], ldsF[0], 0, tid);
    wait_async_all();
    __syncthreads();

    for (int kci = 0; kci < NCHUNK; ++kci) {
        const int   kc  = kci * CK;
        const int   buf = kci & 1;
        const float* Vb = ldsV[buf];
        const float* Fb = ldsF[buf];

        // kick off async staging of the next chunk into the other buffer;
        // it streams to LDS while this chunk's 64 WMMAs execute
        if (kci + 1 < NCHUNK)
            stage_chunk_async(V, fcw, ldsV[buf ^ 1], ldsF[buf ^ 1],
                              kc + CK, tid);

        // ---- K inner loop: 16 steps of K=4 ----
        #pragma unroll 4
        for (int k0 = 0; k0 < CK; k0 += 4) {
            const int kk = k0 + 2 * h;
            // A fragment (ISA 16x4 layout): lane m holds K = 2h, 2h+1
            v2f a = *(const v2f*)(xrow + kc + kk);

            // fused per-patch fc_w dots (same x values, zero extra HBM)
            v2f f0 = *(const v2f*)(Fb + 0 * CK + kk);
            v2f f1 = *(const v2f*)(Fb + 1 * CK + kk);
            dp0 += a.x * f0.x + a.y * f0.y;
            dp1 += a.x * f1.x + a.y * f1.y;

            // B fragments (4x16 layout mirroring A): col = c*16 + m
            v2f b0 = *(const v2f*)(Vb + (0 * 16 + m) * CKP + kk);
            v2f b1 = *(const v2f*)(Vb + (1 * 16 + m) * CKP + kk);
            v2f b2 = *(const v2f*)(Vb + (2 * 16 + m) * CKP + kk);
            v2f b3 = *(const v2f*)(Vb + (3 * 16 + m) * CKP + kk);

            acc0 = __builtin_amdgcn_wmma_f32_16x16x4_f32(false, a, false, b0,
                                                         (short)0, acc0, false, false);
            acc1 = __builtin_amdgcn_wmma_f32_16x16x4_f32(false, a, false, b1,
                                                         (short)0, acc1, false, false);
            acc2 = __builtin_amdgcn_wmma_f32_16x16x4_f32(false, a, false, b2,
                                                         (short)0, acc2, false, false);
            acc3 = __builtin_amdgcn_wmma_f32_16x16x4_f32(false, a, false, b3,
                                                         (short)0, acc3, false, false);
        }

        // make next buffer's staging visible to all waves before reuse
        wait_async_all();
        __syncthreads();
    }

    // ---- epilogue: dis[row] = sum_col tanh(C[row,col]) * W[col] / 8 ----
    // C layout: VGPR r, lanes 0-15 -> (M=r, N=lane); lanes 16-31 -> (M=r+8).
    const float w0 = W[0 * 16 + m];
    const float w1 = W[1 * 16 + m];
    const float w2 = W[2 * 16 + m];
    const float w3 = W[3 * 16 + m];

    float pd[8];
    #pragma unroll
    for (int r = 0; r < 8; ++r) {
        pd[r] = tanhf(acc0[r]) * w0 + tanhf(acc1[r]) * w1 +
                tanhf(acc2[r]) * w2 + tanhf(acc3[r]) * w3;
    }
    // reduce over the 16 columns in each wave32 half (masks < 16 stay in-half)
    #pragma unroll
    for (int s = 8; s >= 1; s >>= 1) {
        #pragma unroll
        for (int r = 0; r < 8; ++r)
            pd[r] += __shfl_xor(pd[r], s, 32);
    }
    if (m < 8)                                     // lanes 0-7 -> rows 0-7, 16-23 -> rows 8-15
        dis[rowbase + m + 8 * h] = pd[m] * 0.125f; // 1/sqrt(DL)

    // ---- fc dots: row m lives in lanes m and m+16 (the two K halves) ----
    dp0 += __shfl_xor(dp0, 16, 32);
    dp1 += __shfl_xor(dp1, 16, 32);
    if (h == 0) {
        dotc[(rowbase + m) * NCLS_ + 0] = dp0;
        dotc[(rowbase + m) * NCLS_ + 1] = dp1;
    }
}

// ---------------------------------------------------------------------------
// Kernel 2: masked softmax over N for one bag per block.
// ---------------------------------------------------------------------------
__global__ __launch_bounds__(256) void wsinet_softmax(
    const float* __restrict__ dis, const int* __restrict__ true_num,
    float* __restrict__ soft)       // [B, N] (lives at d_out + B*NCLS)
{
    __shared__ float sv[N_];        // 16 KB
    __shared__ float red[256];
    const int b = blockIdx.x, tid = threadIdx.x;
    const int tn = true_num[b];

    float lmax = -3.0e38f;
    for (int n = tid; n < N_; n += 256) {
        float v = dis[b * N_ + n];
        v = (n < tn) ? v : -1.0e20f;
        sv[n] = v;
        lmax = fmaxf(lmax, v);
    }
    red[tid] = lmax;
    __syncthreads();
    for (int s = 128; s > 0; s >>= 1) {
        if (tid < s) red[tid] = fmaxf(red[tid], red[tid + s]);
        __syncthreads();
    }
    const float mx = red[0];
    __syncthreads();

    float lsum = 0.0f;
    for (int n = tid; n < N_; n += 256) {
        float e = __expf(sv[n] - mx);
        sv[n] = e;
        lsum += e;
    }
    red[tid] = lsum;
    __syncthreads();
    for (int s = 128; s > 0; s >>= 1) {
        if (tid < s) red[tid] += red[tid + s];
        __syncthreads();
    }
    const float inv = 1.0f / red[0];
    for (int n = tid; n < N_; n += 256)
        soft[b * N_ + n] = sv[n] * inv;
}

// ---------------------------------------------------------------------------
// Kernel 3: logits[b,c] = sum_n s[b,n]*dotc[b,n,c] + fc_b[c]; class softmax.
// ---------------------------------------------------------------------------
__global__ __launch_bounds__(256) void wsinet_head(
    const float* __restrict__ dotc, const float* __restrict__ soft,
    const float* __restrict__ fcb, float* __restrict__ cls)
{
    __shared__ float r0[256], r1[256];
    const int b = blockIdx.x, tid = threadIdx.x;
    float p0 = 0.0f, p1 = 0.0f;
    for (int n = tid; n < N_; n += 256) {
        float s = soft[b * N_ + n];
        p0 += s * dotc[(b * N_ + n) * NCLS_ + 0];
        p1 += s * dotc[(b * N_ + n) * NCLS_ + 1];
    }
    r0[tid] = p0; r1[tid] = p1;
    __syncthreads();
    for (int s = 128; s > 0; s >>= 1) {
        if (tid < s) { r0[tid] += r0[tid + s]; r1[tid] += r1[tid + s]; }
        __syncthreads();
    }
    if (tid == 0) {
        float l0 = r0[0] + fcb[0];
        float l1 = r1[0] + fcb[1];
        float mm = fmaxf(l0, l1);
        float e0 = __expf(l0 - mm), e1 = __expf(l1 - mm);
        float inv = 1.0f / (e0 + e1);
        cls[b * NCLS_ + 0] = e0 * inv;
        cls[b * NCLS_ + 1] = e1 * inv;
    }
}

// ---------------------------------------------------------------------------
extern "C" void kernel_launch(void* const* d_in, const int* in_sizes, int n_in,
                              void* d_out, int out_size, void* d_ws, size_t ws_size,
                              hipStream_t stream)
{
    const float* x   = (const float*)d_in[0];   // [B,N,D]
    const int*   tn  = (const int*)  d_in[1];   // [B]
    const float* V   = (const float*)d_in[2];   // [D,DL]
    const float* W   = (const float*)d_in[3];   // [DL,1]
    const float* fcw = (const float*)d_in[4];   // [NCLS,D]
    const float* fcb = (const float*)d_in[5];   // [NCLS]

    float* out  = (float*)d_out;
    float* cls  = out;                 // B*NCLS
    float* soft = out + B_ * NCLS_;    // B*N

    float* dis  = (float*)d_ws;                    // B*N floats
    float* dotc = dis + (size_t)B_ * N_;           // B*N*NCLS floats (384KB total)

    wsinet_scores<<<(B_ * N_) / 128, 256, 0, stream>>>(x, V, W, fcw, dis, dotc);
    wsinet_softmax<<<B_, 256, 0, stream>>>(dis, tn, soft);
    wsinet_head<<<B_, 256, 0, stream>>>(dotc, soft, fcb, cls);
}